// EmanAttLayer_13005160972937
// MI455X (gfx1250) — compile-verified
//
#include <hip/hip_runtime.h>
#include <hip/hip_bf16.h>

// ---------------- problem constants (from reference) ----------------
#define C_IN   16
#define C_OUT  16
#define MDIM   5     // 2*L_IN+1
#define NDIM   5     // 2*L_OUT+1
#define RDIM   4
#define FDIM   5
#define BN     10    // neigh basis
#define BS     5     // self basis
#define KN_TOT (BN*RDIM*C_IN)   // 640  (b,r,j) flattened K for neigh GEMM
#define KS_TOT (BS*RDIM*C_IN)   // 320
#define KN_STR (KN_TOT + 4)     // 644  padded row stride (conflict-free b64 loads)
#define KS_STR (KS_TOT + 4)     // 324
#define ZN_WORDS (NDIM * KN_STR)   // 3220 per edge
#define ZS_WORDS (NDIM * KS_STR)   // 1620 per edge
#define EPW    3     // edges per wave (3 x 5 irrep columns = 15 of 16 WMMA cols)
#define NWAVES 1
#define THREADS (NWAVES*32)

typedef float v2f __attribute__((ext_vector_type(2)));
typedef float v8f __attribute__((ext_vector_type(8)));

#define LDS_FENCE() asm volatile("s_wait_dscnt 0" ::: "memory")

// --------------------------------------------------------------------
// Main kernel: one wave32 handles EPW=3 edges at a time.
//   y_dst = W_n[16 x 640] @ z_n[640 x 15]  (fp32 WMMA 16x16x4, cols = 3 edges x 5)
//   y_src = W_s[16 x 320] @ z_s[320 x 15]
//   prods[e,h] = sum_head(y_src*y_dst) / sqrt(40)
// --------------------------------------------------------------------
__global__ __launch_bounds__(THREADS)
void eman_main(const float* __restrict__ x,       // [N_NODES][16][5]
               const int*   __restrict__ ei,      // [2][E]
               const float* __restrict__ pcn,     // [E][5][4]
               const float* __restrict__ pcs,     // [E][5][4]
               const float* __restrict__ conn,    // [E]
               const float* __restrict__ kn,      // [10][5][5][5]
               const float* __restrict__ ksf,     // [5][5][5][5]
               const float* __restrict__ wn,      // [10][4][16][16]
               const float* __restrict__ wsf,     // [5][4][16][16]
               float* __restrict__ prods,         // [E][2]
               int E)
{
    // shared (edge-independent), transposed+padded: W[i][k], row stride K+4
    __shared__ __align__(16) float sWnT[16 * KN_STR];           // 41216 B
    __shared__ __align__(16) float sWsT[16 * KS_STR];           // 20736 B
    __shared__ __align__(16) float sKn[BN*FDIM*NDIM*MDIM];      // 5000 B
    __shared__ __align__(16) float sKs[BS*FDIM*NDIM*MDIM];      // 2500 B
    // per-wave, per-edge scratch
    __shared__ __align__(16) float sXt[NWAVES][EPW][C_IN*MDIM]; // transported x_j
    __shared__ __align__(16) float sXi[NWAVES][EPW][C_IN*MDIM]; // x_i
    __shared__ __align__(16) float sPn[NWAVES][EPW][FDIM*RDIM];
    __shared__ __align__(16) float sPs[NWAVES][EPW][FDIM*RDIM];
    __shared__ __align__(16) float sQ [NWAVES][EPW][BN*RDIM*NDIM*MDIM]; // 1000f, reused for self (500f)
    __shared__ __align__(16) float sZn[NWAVES][EPW*ZN_WORDS];   // z_n^T [n][k], 38640 B
    __shared__ __align__(16) float sZs[NWAVES][EPW*ZS_WORDS];   // z_s^T [n][k], 19440 B
    __shared__ __align__(16) float sRed[NWAVES][32];

    const int tid  = threadIdx.x;
    const int lane = tid & 31;
    const int wave = tid >> 5;

    // ---- cooperative load of shared operands (transposed weights) ----
    for (int t = tid; t < KN_TOT*16; t += THREADS) {
        int i = t & 15, k = t >> 4;
        sWnT[i*KN_STR + k] = wn[(k >> 4)*256 + i*16 + (k & 15)];
    }
    for (int t = tid; t < KS_TOT*16; t += THREADS) {
        int i = t & 15, k = t >> 4;
        sWsT[i*KS_STR + k] = wsf[(k >> 4)*256 + i*16 + (k & 15)];
    }
    for (int t = tid; t < BN*FDIM*NDIM*MDIM; t += THREADS) sKn[t] = kn[t];
    for (int t = tid; t < BS*FDIM*NDIM*MDIM; t += THREADS) sKs[t] = ksf[t];
    LDS_FENCE();

    const int hlf  = lane >> 4;          // C/D half: rows 0-7 (head 0) / 8-15 (head 1)
    const int col  = lane & 15;          // D column; also A row index i
    const int gcl  = (col >= 10) ? 2 : (col >= 5 ? 1 : 0);  // edge within group (col 15 -> 2)
    const int colc = col - ((col / 5 > 2 ? 2 : col / 5) * 5) % 5 ? col % 5 : col % 5; // = col % 5
    const int cmod = col % 5;

    // per-lane constant fragment pointers (v2f units; all offsets even)
    const v2f* aVn = ((const v2f*)sWnT) + col*(KN_STR/2) + hlf;
    const v2f* aVs = ((const v2f*)sWsT) + col*(KS_STR/2) + hlf;
    const v2f* bVn = ((const v2f*)sZn[wave]) + gcl*(ZN_WORDS/2) + cmod*(KN_STR/2) + hlf;
    const v2f* bVs = ((const v2f*)sZs[wave]) + gcl*(ZS_WORDS/2) + cmod*(KS_STR/2) + hlf;
    (void)colc;

    const int wid = blockIdx.x * NWAVES + wave;
    const int stride3 = gridDim.x * NWAVES * EPW;

    for (int base = wid * EPW; base < E; base += stride3) {
        // ---- features: lanes 0-15 transport x[dst], lanes 16-31 copy x[src] ----
#pragma unroll
        for (int g = 0; g < EPW; ++g) {
            int eC = base + g;  if (eC >= E) eC = E - 1;
            const int srcN = ei[eC];
            const int dstN = ei[E + eC];
            const float theta = conn[eC];
            const int j = col;
            const float* xp = x + (lane < 16 ? dstN : srcN)*80 + j*MDIM;
            float v0 = xp[0], v1 = xp[1], v2 = xp[2], v3 = xp[3], v4 = xp[4];
            if (lane < 16) {
                float c1 = __cosf(theta),      s1 = __sinf(theta);
                float c2 = __cosf(2.0f*theta), s2 = __sinf(2.0f*theta);
                float* o = &sXt[wave][g][j*MDIM];
                o[0] = v0;
                o[1] = c1*v1 - s1*v2;  o[2] = s1*v1 + c1*v2;
                o[3] = c2*v3 - s2*v4;  o[4] = s2*v3 + c2*v4;
            } else {
                float* o = &sXi[wave][g][j*MDIM];
                o[0] = v0; o[1] = v1; o[2] = v2; o[3] = v3; o[4] = v4;
            }
            if (lane < FDIM*RDIM) {
                sPn[wave][g][lane] = pcn[eC*20 + lane];
                sPs[wave][g][lane] = pcs[eC*20 + lane];
            }
        }
        LDS_FENCE();

        // ---- q_n[b,r,n,m] = sum_f Kn[b,f,n,m] * p_n[f,r] ----
        for (int g = 0; g < EPW; ++g)
            for (int t = lane; t < BN*RDIM*NDIM*MDIM; t += 32) {
                int m = t % 5, n = (t/5) % 5, br = t / 25;
                int r = br & 3, b = br >> 2;
                const float* kb = &sKn[b*125 + n*5 + m];
                const float* pr = &sPn[wave][g][r];
                float a = 0.f;
#pragma unroll
                for (int f = 0; f < FDIM; ++f) a += kb[f*25] * pr[f*4];
                sQ[wave][g][t] = a;
            }
        LDS_FENCE();

        // ---- z_n^T[n][k=(b,r,j)] = sum_m q_n[b,r,n,m] * xt[j,m] ----
        for (int g = 0; g < EPW; ++g)
            for (int t = lane; t < KN_TOT*NDIM; t += 32) {
                int n = t % 5, k = t / 5, j = k & 15, br = k >> 4;
                const float* q  = &sQ[wave][g][(br*5 + n)*5];
                const float* xv = &sXt[wave][g][j*5];
                sZn[wave][g*ZN_WORDS + n*KN_STR + k] =
                    q[0]*xv[0] + q[1]*xv[1] + q[2]*xv[2] + q[3]*xv[3] + q[4]*xv[4];
            }
        LDS_FENCE();

        // ---- q_s (reuses sQ buffer) ----
        for (int g = 0; g < EPW; ++g)
            for (int t = lane; t < BS*RDIM*NDIM*MDIM; t += 32) {
                int m = t % 5, n = (t/5) % 5, br = t / 25;
                int r = br & 3, b = br >> 2;
                const float* kb = &sKs[b*125 + n*5 + m];
                const float* pr = &sPs[wave][g][r];
                float a = 0.f;
#pragma unroll
                for (int f = 0; f < FDIM; ++f) a += kb[f*25] * pr[f*4];
                sQ[wave][g][t] = a;
            }
        LDS_FENCE();

        // ---- z_s^T ----
        for (int g = 0; g < EPW; ++g)
            for (int t = lane; t < KS_TOT*NDIM; t += 32) {
                int n = t % 5, k = t / 5, j = k & 15, br = k >> 4;
                const float* q  = &sQ[wave][g][(br*5 + n)*5];
                const float* xv = &sXi[wave][g][j*5];
                sZs[wave][g*ZS_WORDS + n*KS_STR + k] =
                    q[0]*xv[0] + q[1]*xv[1] + q[2]*xv[2] + q[3]*xv[3] + q[4]*xv[4];
            }
        LDS_FENCE();

        // ---- interleaved WMMA GEMMs: two independent accumulator chains ----
        v8f accN = {0.f,0.f,0.f,0.f,0.f,0.f,0.f,0.f};
        v8f accS = {0.f,0.f,0.f,0.f,0.f,0.f,0.f,0.f};
#pragma unroll 4
        for (int k0 = 0; k0 < KS_TOT; k0 += 4) {          // 80 iters: both paths
            v2f an = aVn[k0 >> 1], bn = bVn[k0 >> 1];
            v2f as = aVs[k0 >> 1], bs = bVs[k0 >> 1];
            accN = __builtin_amdgcn_wmma_f32_16x16x4_f32(
                false, an, false, bn, (short)0, accN, false, false);
            accS = __builtin_amdgcn_wmma_f32_16x16x4_f32(
                false, as, false, bs, (short)0, accS, false, false);
        }
#pragma unroll 4
        for (int k0 = KS_TOT; k0 < KN_TOT; k0 += 4) {     // 80 iters: neigh tail
            v2f an = aVn[k0 >> 1], bn = bVn[k0 >> 1];
            accN = __builtin_amdgcn_wmma_f32_16x16x4_f32(
                false, an, false, bn, (short)0, accN, false, false);
        }

        // ---- per-(edge,head) dot product: D col = 5*g + n, head = hlf ----
        float s = 0.f;
#pragma unroll
        for (int c = 0; c < 8; ++c) s += accN[c] * accS[c];
        sRed[wave][lane] = s;
        LDS_FENCE();
        if (cmod == 0 && col < 15) {                      // cols 0,5,10 per half
            const float* rp = &sRed[wave][hlf*16 + col];
            float sum = rp[0] + rp[1] + rp[2] + rp[3] + rp[4];
            int eg = base + col / 5;
            if (eg < E) prods[eg*2 + hlf] = sum * 0.15811388300841897f; // 1/sqrt(40)
        }
        LDS_FENCE();
    }
}

// ------------------- softmax epilogue kernels -----------------------
__device__ __forceinline__ unsigned fenc(float f) {
    unsigned u = __float_as_uint(f);
    return u ^ ((u >> 31) ? 0xFFFFFFFFu : 0x80000000u);
}
__device__ __forceinline__ float fdec(unsigned u) {
    return (u & 0x80000000u) ? __uint_as_float(u ^ 0x80000000u) : __uint_as_float(~u);
}

__global__ void k_init(int* deg, unsigned* mkey, float* den, int NN) {
    int i = blockIdx.x * blockDim.x + threadIdx.x;
    if (i < NN) deg[i] = 0;
    if (i < 2*NN) { mkey[i] = 0u; den[i] = 0.f; }
}

__global__ void k_degmax(const int* __restrict__ ei, const float* __restrict__ prods,
                         int* deg, unsigned* mkey, int E) {
    int e = blockIdx.x * blockDim.x + threadIdx.x;
    if (e >= E) return;
    int s = ei[e];
    atomicAdd(&deg[s], 1);
    atomicMax(&mkey[s*2 + 0], fenc(prods[e*2 + 0]));
    atomicMax(&mkey[s*2 + 1], fenc(prods[e*2 + 1]));
}

__global__ void k_den(const int* __restrict__ ei, const float* __restrict__ prods,
                      const unsigned* __restrict__ mkey, float* den, int E) {
    int e = blockIdx.x * blockDim.x + threadIdx.x;
    if (e >= E) return;
    int s = ei[e];
#pragma unroll
    for (int h = 0; h < 2; ++h) {
        float m = fdec(mkey[s*2 + h]);
        atomicAdd(&den[s*2 + h], __expf(prods[e*2 + h] - m));
    }
}

__global__ void k_final(const int* __restrict__ ei, const float* __restrict__ prods,
                        const unsigned* __restrict__ mkey, const float* __restrict__ den,
                        const int* __restrict__ deg, float* __restrict__ out, int E) {
    int e = blockIdx.x * blockDim.x + threadIdx.x;
    if (e >= E) return;
    int s = ei[e];
    float d = (float)deg[s];
#pragma unroll
    for (int h = 0; h < 2; ++h) {
        float m = fdec(mkey[s*2 + h]);
        out[e*2 + h] = d * __expf(prods[e*2 + h] - m) / den[s*2 + h];
    }
}

// --------------------------------------------------------------------
extern "C" void kernel_launch(void* const* d_in, const int* in_sizes, int n_in,
                              void* d_out, int out_size, void* d_ws, size_t ws_size,
                              hipStream_t stream) {
    const float* x    = (const float*)d_in[0];
    const int*   ei   = (const int*)  d_in[1];
    const float* pcn  = (const float*)d_in[2];
    const float* pcs  = (const float*)d_in[3];
    const float* conn = (const float*)d_in[4];
    const float* kn   = (const float*)d_in[5];
    const float* ksf  = (const float*)d_in[6];
    const float* wn   = (const float*)d_in[7];
    const float* wsf  = (const float*)d_in[8];
    float* out = (float*)d_out;

    const int E  = in_sizes[4];                 // 160000 edges
    const int NN = in_sizes[0] / (C_IN * MDIM); // 10000 nodes

    char* ws = (char*)d_ws;
    float*    prods = (float*)   ws;                                  // E*2 floats
    int*      deg   = (int*)     (ws + (size_t)E*2*4);                // NN ints
    unsigned* mkey  = (unsigned*)(ws + (size_t)E*2*4 + (size_t)NN*4); // 2*NN
    float*    den   = (float*)   (ws + (size_t)E*2*4 + (size_t)NN*4 + (size_t)NN*8);

    k_init<<<(2*NN + 255)/256, 256, 0, stream>>>(deg, mkey, den, NN);

    int ngroups = (E + EPW - 1) / EPW;          // wave-iterations needed
    int mgrid = ngroups < 8192 ? ngroups : 8192;
    eman_main<<<mgrid, THREADS, 0, stream>>>(x, ei, pcn, pcs, conn, kn, ksf, wn, wsf,
                                             prods, E);

    int eg = (E + 255)/256;
    k_degmax<<<eg, 256, 0, stream>>>(ei, prods, deg, mkey, E);
    k_den   <<<eg, 256, 0, stream>>>(ei, prods, mkey, den, E);
    k_final <<<eg, 256, 0, stream>>>(ei, prods, mkey, den, deg, out, E);
}